// CorrKernel_29798483099806
// MI455X (gfx1250) — compile-verified
//
#include <hip/hip_runtime.h>

// Problem constants (from reference): B=8, C=256, H=64, W=128, PATCH=9, L=81
#define BB   8
#define CC   256
#define HH   64
#define WW   128
#define PATCHN 9
#define LL   81
#define KC   8              // channels staged per LDS chunk
#define NKC  (CC / KC)      // 32 chunks
#define F1W  144            // padded LDS row width for fmap1 (4 left, 12 right zeros)

typedef __attribute__((ext_vector_type(2))) float v2f;
typedef __attribute__((ext_vector_type(4))) float v4f;
typedef __attribute__((ext_vector_type(8))) float v8f;
typedef __attribute__((ext_vector_type(4))) int   v4i;

// ---- async global->LDS staging (CDNA5 ASYNCcnt path), with safe fallback ----
#if defined(__gfx1250__) && __has_builtin(__builtin_amdgcn_global_load_async_to_lds_b128)
#define HAS_ASYNC 1
__device__ __forceinline__ void async_copy16(const float* g, float* l) {
    __builtin_amdgcn_global_load_async_to_lds_b128(
        (__attribute__((address_space(1))) v4i*)(g),   // global src (AS1), non-const
        (__attribute__((address_space(3))) v4i*)(l),   // LDS dst (AS3)
        /*offset=*/0, /*cpol=*/0);
}
#if __has_builtin(__builtin_amdgcn_s_wait_asynccnt)
#define ASYNC_WAIT() __builtin_amdgcn_s_wait_asynccnt(0)
#else
#define ASYNC_WAIT() asm volatile("s_wait_asynccnt 0" ::: "memory")
#endif
#else
#define HAS_ASYNC 0
__device__ __forceinline__ void async_copy16(const float* g, float* l) {
    *(v4f*)l = *(const v4f*)g;
}
#define ASYNC_WAIT() ((void)0)
#endif

// One workgroup per (b, h). Wave t (0..7) owns output pixels w in [16t, 16t+16).
// For each dy, the 9-dx band of correlations is the band of the pair-product
// R[w, w'] = sum_c f0[c,w] * f1[c,h+dy-4,w'], computed with two
// V_WMMA_F32_16X16X4_F32 N-tiles per k-step. corr[m,dx] = R[m, m+dx].
__global__ __launch_bounds__(256)
void corr_wmma_kernel(const float* __restrict__ f0,
                      const float* __restrict__ f1,
                      float* __restrict__ out)
{
    __shared__ __align__(16) float smem[KC * WW + PATCHN * KC * F1W]; // 45,568 B
    float* s_f0 = smem;                 // [KC][WW]
    float* s_f1 = smem + KC * WW;       // [PATCHN][KC][F1W], col = w' + 4
    float* s_R  = smem;                 // alias after compute: [8 tiles][16 m][32 n]

    const int tid  = threadIdx.x;
    const int lane = tid & 31;
    const int wv   = tid >> 5;          // wave id == w-tile index t
    const int lh   = lane >> 4;         // lane half (0/1)
    const int lm   = lane & 15;

    const int bh = blockIdx.x;          // 0 .. B*H-1
    const int b  = bh >> 6;             // / H(=64)
    const int h  = bh & (HH - 1);

    const float* f0row  = f0 + (((size_t)b * CC) * HH + h) * WW;   // + c*H*W + w
    const float* f1base = f1 + (size_t)b * CC * HH * WW;

    // Zero the f1 stage once: padding columns and OOB-dy rows stay zero forever
    // (per-chunk async loads only rewrite valid interiors).
    for (int i = tid; i < PATCHN * KC * F1W; i += 256) s_f1[i] = 0.0f;
    __syncthreads();

    v8f acc[2 * PATCHN];
    #pragma unroll
    for (int i = 0; i < 2 * PATCHN; ++i)
        acc[i] = v8f{0.f, 0.f, 0.f, 0.f, 0.f, 0.f, 0.f, 0.f};

    const int kstage = tid >> 5;            // 0..7 channel within chunk
    const int wq     = (tid & 31) << 2;     // 16B-aligned w offset 0,4,...,124

    for (int kc = 0; kc < NKC; ++kc) {
        const int c0 = kc * KC;

        // ---- stage f0 chunk: 8 channels x 128 floats (1x b128 per thread)
        async_copy16(f0row + (size_t)(c0 + kstage) * HH * WW + wq,
                     s_f0 + kstage * WW + wq);

        // ---- stage f1 chunk: 9 rows x 8 channels x 128 floats (9x b128/thread)
        #pragma unroll
        for (int r = 0; r < PATCHN; ++r) {
            const int h1 = h + r - 4;
            if (h1 < 0 || h1 >= HH) continue;   // uniform per block: stays zero
            async_copy16(f1base + ((size_t)(c0 + kstage) * HH + h1) * WW + wq,
                         s_f1 + (r * KC + kstage) * F1W + 4 + wq);
        }
        ASYNC_WAIT();
        __syncthreads();

        // ---- WMMA accumulation over this channel chunk
        #pragma unroll
        for (int ks = 0; ks < KC / 4; ++ks) {
            const int kk = ks * 4 + 2 * lh;     // lane-half picks K pair
            // A (16x4): lanes 0-15 hold K=k0,k0+1 ; lanes 16-31 hold K=k0+2,k0+3
            const float* pa = s_f0 + kk * WW + (wv << 4) + lm;
            v2f a; a.x = pa[0]; a.y = pa[WW];
            #pragma unroll
            for (int dy = 0; dy < PATCHN; ++dy) {
                const float* pb = s_f1 + (dy * KC + kk) * F1W + (wv << 4) + lm;
                v2f bl; bl.x = pb[0];        bl.y = pb[F1W];        // N-tile left
                v2f br; br.x = pb[16];       br.y = pb[16 + F1W];   // N-tile right
                acc[dy] = __builtin_amdgcn_wmma_f32_16x16x4_f32(
                    false, a, false, bl, (short)0, acc[dy], false, false);
                acc[PATCHN + dy] = __builtin_amdgcn_wmma_f32_16x16x4_f32(
                    false, a, false, br, (short)0, acc[PATCHN + dy], false, false);
            }
        }
        __syncthreads();   // WAR: LDS reads done before next chunk's stores
    }

    // ---- band extraction + coalesced writeout: corr[m,dx] = R[m, m+dx]
    float* outb = out + (size_t)b * LL * HH * WW + (size_t)h * WW;
    #pragma unroll
    for (int dy = 0; dy < PATCHN; ++dy) {
        // C/D layout: VGPR r, lane<16 -> (M=r, N=lane); lane>=16 -> (M=r+8, N=lane-16)
        #pragma unroll
        for (int r = 0; r < 8; ++r) {
            const int m = r + (lh << 3);
            s_R[((wv * 16 + m) << 5) + lm]      = acc[dy][r];
            s_R[((wv * 16 + m) << 5) + 16 + lm] = acc[PATCHN + dy][r];
        }
        __syncthreads();
        for (int idx = tid; idx < PATCHN * WW; idx += 256) {
            const int dx = idx >> 7;            // 0..8
            const int w  = idx & (WW - 1);
            const int t2 = w >> 4, m2 = w & 15;
            const float v = s_R[((t2 * 16 + m2) << 5) + m2 + dx];
            outb[(size_t)(dy * PATCHN + dx) * HH * WW + w] = v;
        }
        __syncthreads();
    }
}

extern "C" void kernel_launch(void* const* d_in, const int* in_sizes, int n_in,
                              void* d_out, int out_size, void* d_ws, size_t ws_size,
                              hipStream_t stream) {
    const float* f0 = (const float*)d_in[0];
    const float* f1 = (const float*)d_in[1];
    // d_in[2]/d_in[3] (ox, oy) are the fixed 9x9 meshgrid; the kernel enumerates
    // l = dy*9 + dx in the same row-major ('ij') order as the reference.
    float* out = (float*)d_out;
    corr_wmma_kernel<<<dim3(BB * HH), dim3(256), 0, stream>>>(f0, f1, out);
}